// NormDistConv_944892805269
// MI455X (gfx1250) — compile-verified
//
#include <hip/hip_runtime.h>
#include <stdint.h>

// NormDistConv: out[n,o,l] = max_f |im2col(x)[n,f,l] - W[o,f]| + bias[o]
// N=8, C=16, H=W=32, 3x3 pad=1 stride=1 -> F=144, O=64, L=1024.
// Max-plus op => WMMA inapplicable; VALU-bound (151M lane-ops vs 2.5MB traffic).
// CDNA5 path: TDM (tensor_load_to_lds) stages weights to LDS, s_wait_tensorcnt,
// aligned ds_load_b128 weight reads, wave32 blocks.

namespace {
constexpr int kInCh  = 16;
constexpr int kOutCh = 64;
constexpr int kH = 32, kW = 32;
constexpr int kF     = kInCh * 9;   // 144
constexpr int kOBlk  = 16;          // output channels per block (gridDim.y = 4)
constexpr int kPixBlk = 256;        // pixels per block == blockDim.x (8 wave32)
constexpr int kSlot  = 12;          // 9 weights padded to 12 floats -> 48B, 16B aligned
}

typedef unsigned int v4u __attribute__((ext_vector_type(4)));
typedef int          v8i __attribute__((ext_vector_type(8)));
typedef int          v4i __attribute__((ext_vector_type(4)));

__global__ __launch_bounds__(kPixBlk) void cheb_conv3x3(
    const float* __restrict__ x, const float* __restrict__ weight,
    const float* __restrict__ bias, float* __restrict__ out)
{
  const int tid = threadIdx.x;
  const int n   = blockIdx.x >> 2;                    // image
  const int l   = ((blockIdx.x & 3) * kPixBlk) + tid; // output pixel in [0,1024)
  const int o0  = blockIdx.y * kOBlk;                 // first output channel

  __shared__ float stage[kOBlk * kF];                 // raw TDM landing zone (9216 B)
  __shared__ float wl[kInCh * kOBlk * kSlot];         // [c][o][12] padded layout (12288 B)

#if __has_builtin(__builtin_amdgcn_tensor_load_to_lds) && __has_builtin(__builtin_amdgcn_s_wait_tensorcnt)
  if (tid < 32) {  // one wave issues the TDM op (uniform branch; TDM ignores EXEC)
    const uint64_t ga  = (uint64_t)(uintptr_t)(weight + (size_t)o0 * kF);
    const uint32_t lds = (uint32_t)(uintptr_t)(&stage[0]); // low 32 bits of flat LDS addr = LDS offset
    // D# group0: count=1 | lds_addr | global_addr[56:0] | type=2 ("image")
    v4u g0 = { 1u, lds, (uint32_t)ga,
               (uint32_t)((ga >> 32) & 0x1FFFFFFull) | (2u << 30) };
    // D# group1: data_size=4B(code 2); tensor_dim0=144; tensor_dim1=16;
    //            tile_dim0=144; tile_dim1=16; tensor_dim0_stride=144.
    v8i g1 = { (int)(2u << 16),                 // workgroup_mask=0 | data_size=2
               (int)((uint32_t)kF    << 16),    // barrier_addr=0 | tensor_dim0.lo16
               (int)((uint32_t)kOBlk << 16),    // tensor_dim0.hi16=0 | tensor_dim1.lo16
               (int)((uint32_t)kF    << 16),    // tensor_dim1.hi16=0 | tile_dim0
               (int)kOBlk,                      // tile_dim1 | tile_dim2=0
               (int)kF,                         // tensor_dim0_stride.lo32
               0, 0 };
    v4i gz4 = { 0, 0, 0, 0 };                   // groups 2/3 unused (2D tensor)
    v8i gz8 = { 0, 0, 0, 0, 0, 0, 0, 0 };       // extra group (clang-23 6-arg form)
    __builtin_amdgcn_tensor_load_to_lds(g0, g1, gz4, gz4, gz8, 0);
    __builtin_amdgcn_s_wait_tensorcnt(0);       // data in LDS before we hit the barrier
  }
  __syncthreads();
  for (int idx = tid; idx < kOBlk * kF; idx += kPixBlk) {
    const int o = idx / kF, f = idx - o * kF;
    const int c = f / 9,   k = f - c * 9;
    wl[(c * kOBlk + o) * kSlot + k] = stage[idx];
  }
#else
  for (int idx = tid; idx < kOBlk * kF; idx += kPixBlk) {
    const int o = idx / kF, f = idx - o * kF;
    const int c = f / 9,   k = f - c * 9;
    wl[(c * kOBlk + o) * kSlot + k] = weight[(size_t)(o0 + o) * kF + f];
  }
  (void)stage;
#endif
  __syncthreads();

  float dist[kOBlk];
#pragma unroll
  for (int o = 0; o < kOBlk; ++o) dist[o] = 0.0f;   // |.| >= 0, so 0-init is exact

  const int ho = l >> 5, wo = l & 31;
  const float* xn = x + (size_t)n * kInCh * kH * kW;
  __builtin_prefetch(xn + l, 0, 0);                 // global_prefetch_b8

  for (int c = 0; c < kInCh; ++c) {
    const float* xc = xn + c * (kH * kW);
    float p[9];
#pragma unroll
    for (int i = 0; i < 3; ++i)
#pragma unroll
      for (int j = 0; j < 3; ++j) {
        const int hh = ho + i - 1, ww = wo + j - 1;
        const bool ok = (hh >= 0) && (hh < kH) && (ww >= 0) && (ww < kW);
        p[i * 3 + j] = ok ? xc[hh * kW + ww] : 0.0f; // zero padding
      }
#pragma unroll
    for (int o = 0; o < kOBlk; ++o) {
      const float* wp = &wl[(c * kOBlk + o) * kSlot]; // 48B-aligned slot
      const float4 wa = *(const float4*)(wp);         // ds_load_b128
      const float4 wb = *(const float4*)(wp + 4);     // ds_load_b128
      const float  w8 = wp[8];                        // ds_load_b32
      const float a0 = fabsf(p[0] - wa.x);
      const float a1 = fabsf(p[1] - wa.y);
      const float a2 = fabsf(p[2] - wa.z);
      const float a3 = fabsf(p[3] - wa.w);
      const float a4 = fabsf(p[4] - wb.x);
      const float a5 = fabsf(p[5] - wb.y);
      const float a6 = fabsf(p[6] - wb.z);
      const float a7 = fabsf(p[7] - wb.w);
      const float a8 = fabsf(p[8] - w8);
      // balanced tree: backend folds into v_max3_num_f32 with |src| modifiers
      const float m01 = fmaxf(a0, a1), m23 = fmaxf(a2, a3);
      const float m45 = fmaxf(a4, a5), m67 = fmaxf(a6, a7);
      const float m   = fmaxf(fmaxf(m01, m23), fmaxf(m45, m67));
      dist[o] = fmaxf(dist[o], fmaxf(m, a8));
    }
  }

  const size_t obase = ((size_t)n * kOutCh + o0) * (size_t)(kH * kW) + (size_t)l;
#pragma unroll
  for (int o = 0; o < kOBlk; ++o)
    out[obase + (size_t)o * (kH * kW)] = dist[o] + bias[o0 + o];
}

extern "C" void kernel_launch(void* const* d_in, const int* in_sizes, int n_in,
                              void* d_out, int out_size, void* d_ws, size_t ws_size,
                              hipStream_t stream) {
  const float* x = (const float*)d_in[0];
  const float* w = (const float*)d_in[1];
  const float* b = (const float*)d_in[2];
  float* out = (float*)d_out;
  const int N = in_sizes[0] / (kInCh * kH * kW);  // 8
  dim3 grid((unsigned)(N * ((kH * kW) / kPixBlk)), kOutCh / kOBlk); // (32, 4)
  cheb_conv3x3<<<grid, kPixBlk, 0, stream>>>(x, w, b, out);
  (void)n_in; (void)out_size; (void)d_ws; (void)ws_size;
}